// SuperpointDiscriminativeLossopt_7146825581142
// MI455X (gfx1250) — compile-verified
//
#include <hip/hip_runtime.h>
#include <math.h>

typedef __attribute__((ext_vector_type(2))) float v2f;
typedef __attribute__((ext_vector_type(8))) float v8f;

static constexpr int   kD          = 32;
static constexpr int   kClasses    = 20;
static constexpr float kDeltaDist  = 1.2f;
static constexpr float kEps        = 1e-8f;
static constexpr int   kMaxS       = 4096;   // LDS privatization size (S == 4096 here)

// accumulator slots: 0=var_sum 1=reg_sum 2=ent_sum 3=hinge_sum 4=mask_sum

__global__ void k_zero(unsigned int* __restrict__ p, long long nwords) {
  long long i = (long long)blockIdx.x * blockDim.x + threadIdx.x;
  long long stride = (long long)gridDim.x * blockDim.x;
  for (; i < nwords; i += stride) p[i] = 0u;
}

// Normalize superpoint features into fn table; accumulate sum of raw row norms (l_reg).
__global__ void k_norm_super(const float* __restrict__ sp, float* __restrict__ fn,
                             float* __restrict__ acc, int S) {
  int i = blockIdx.x * blockDim.x + threadIdx.x;
  if (i >= S) return;
  const float4* row = (const float4*)(sp + (size_t)i * kD);
  float4 v[8];
  float ss = 0.f;
#pragma unroll
  for (int k = 0; k < 8; ++k) {
    v[k] = row[k];
    ss += v[k].x * v[k].x + v[k].y * v[k].y + v[k].z * v[k].z + v[k].w * v[k].w;
  }
  float n = sqrtf(ss);
  float inv = 1.f / fmaxf(n, 1e-12f);
  float4* out = (float4*)(fn + (size_t)i * kD);
#pragma unroll
  for (int k = 0; k < 8; ++k) {
    float4 w = v[k];
    w.x *= inv; w.y *= inv; w.z *= inv; w.w *= inv;
    out[k] = w;
  }
  atomicAdd(&acc[1], n);
}

// Bandwidth-bound pass over all N points: distance + LDS-privatized segment sums
// + global label histogram.
__global__ void k_points(const float* __restrict__ raw, const float* __restrict__ fn,
                         const int* __restrict__ idx, const int* __restrict__ lab,
                         float* __restrict__ sums, unsigned int* __restrict__ cnts,
                         unsigned int* __restrict__ lcnt, long long N, int S) {
  __shared__ float        lsum[kMaxS];
  __shared__ unsigned int lcount[kMaxS];
  for (int s = threadIdx.x; s < S; s += blockDim.x) { lsum[s] = 0.f; lcount[s] = 0u; }
  __syncthreads();

  long long i = (long long)blockIdx.x * blockDim.x + threadIdx.x;
  long long stride = (long long)gridDim.x * blockDim.x;
  for (; i < N; i += stride) {
    // speculative prefetch of next grid-stride row (global_prefetch_b8; OOB is dropped)
    __builtin_prefetch(raw + (i + stride) * kD, 0, 0);
    const float4* rr = (const float4*)(raw + i * kD);
    int sp = idx[i];
    const float4* sr = (const float4*)(fn + (size_t)sp * kD);
    float ssa = 0.f, ssb = 0.f, dot = 0.f;
#pragma unroll
    for (int k = 0; k < 8; ++k) {
      float4 a = rr[k];
      float4 b = sr[k];
      ssa += a.x * a.x + a.y * a.y + a.z * a.z + a.w * a.w;
      ssb += b.x * b.x + b.y * b.y + b.z * b.z + b.w * b.w;
      dot += a.x * b.x + a.y * b.y + a.z * b.z + a.w * b.w;
    }
    float na   = sqrtf(ssa);
    float inva = 1.f / fmaxf(na, 1e-12f);
    float d2   = ssa * inva * inva + ssb - 2.f * dot * inva;  // ||a_n - b_n||^2
    float dist = sqrtf(fmaxf(d2, 0.f) + 1e-12f);
    atomicAdd(&lsum[sp], dist);      // ds_add_f32
    atomicAdd(&lcount[sp], 1u);      // ds_add_u32
    int c = lab[i];
    if (c >= 0) atomicAdd(&lcnt[(size_t)sp * kClasses + c], 1u);
  }

  __syncthreads();
  for (int s = threadIdx.x; s < S; s += blockDim.x) {
    unsigned int c = lcount[s];
    if (c) {
      atomicAdd(&sums[s], lsum[s]);
      atomicAdd(&cnts[s], c);
    }
  }
}

// Per-superpoint: mean distance (l_var term), entropy, argmax label.
__global__ void k_sp_final(const float* __restrict__ sums, const unsigned int* __restrict__ cnts,
                           const unsigned int* __restrict__ lcnt, int* __restrict__ splab,
                           float* __restrict__ acc, int S) {
  int s = blockIdx.x * blockDim.x + threadIdx.x;
  if (s >= S) return;
  float c = (float)cnts[s];
  float v = sums[s] / fmaxf(c, 1.f);
  atomicAdd(&acc[0], fmaxf(v, 0.f));  // relu + sum (mean done at the end)

  const unsigned int* h = lcnt + (size_t)s * kClasses;
  unsigned int tot = 0, best = 0;
  int barg = 0;
#pragma unroll
  for (int k = 0; k < kClasses; ++k) {
    unsigned int hk = h[k];
    tot += hk;
    if (hk > best) { best = hk; barg = k; }  // strict > == first-max, matches argmax
  }
  splab[s] = barg;
  float denom = (float)tot + kEps;
  float ent = 0.f;
#pragma unroll
  for (int k = 0; k < kClasses; ++k) {
    float p = (float)h[k] / denom;
    ent -= p * logf(p + kEps);
  }
  atomicAdd(&acc[2], ent);
}

// WMMA gram: one wave computes a 16x16 tile of fn @ fn^T via 8x V_WMMA_F32_16X16X4_F32,
// then evaluates the hinge/mask reduction directly on the accumulator VGPRs.
__global__ void __launch_bounds__(32)
k_gram_wmma(const float* __restrict__ fn, const int* __restrict__ splab,
            float* __restrict__ acc, int S) {
  const int tiles = S >> 4;
  const int tile  = blockIdx.x;
  const int ti    = tile / tiles;
  const int tj    = tile - ti * tiles;
  const int lane  = threadIdx.x & 31;
  const int m     = lane & 15;
  const int klo   = (lane >> 4) << 1;  // K pair selector per ISA f32 A/B layout

  const float* arow = fn + (size_t)(ti * 16 + m) * kD + klo;
  const float* brow = fn + (size_t)(tj * 16 + m) * kD + klo;

  v8f cacc = {};
#pragma unroll
  for (int kb = 0; kb < kD; kb += 4) {
    v2f a = *(const v2f*)(arow + kb);  // 8-byte aligned (kb+klo even)
    v2f b = *(const v2f*)(brow + kb);
    cacc = __builtin_amdgcn_wmma_f32_16x16x4_f32(false, a, false, b,
                                                 (short)0, cacc, false, false);
  }

  // C/D layout: VGPR r, lane group g -> row ti*16 + r + 8g, col tj*16 + (lane&15)
  const int j  = tj * 16 + m;
  const int lj = splab[j];
  const int g  = lane >> 4;
  float hsum = 0.f, msum = 0.f;
#pragma unroll
  for (int r = 0; r < 8; ++r) {
    int i = ti * 16 + r + 8 * g;
    if (i == j) continue;                 // exclude diagonal
    if (splab[i] != lj) {                 // different-label pair mask
      float gg = cacc[r];
      float cd = sqrtf(fmaxf(2.f - 2.f * gg, 1e-12f));
      float h  = fmaxf(kDeltaDist - cd, 0.f);
      hsum += h * h;
      msum += 1.f;
    }
  }
  for (int off = 16; off; off >>= 1) {
    hsum += __shfl_xor(hsum, off, 32);
    msum += __shfl_xor(msum, off, 32);
  }
  if (lane == 0) {
    atomicAdd(&acc[3], hsum);
    atomicAdd(&acc[4], msum);
  }
}

__global__ void k_total(const float* __restrict__ acc, float* __restrict__ out, int S) {
  if (blockIdx.x == 0 && threadIdx.x == 0) {
    float fS     = (float)S;
    float l_var  = acc[0] / fS;
    float l_dist = acc[3] / fmaxf(acc[4], 1.f);
    float l_reg  = acc[1] / fS;
    float l_ent  = acc[2] / fS;
    out[0] = 0.1f * (1.0f * l_var + 1.0f * l_dist + 0.001f * l_reg + 1.0f * l_ent);
  }
}

extern "C" void kernel_launch(void* const* d_in, const int* in_sizes, int n_in,
                              void* d_out, int out_size, void* d_ws, size_t ws_size,
                              hipStream_t stream) {
  (void)n_in; (void)out_size; (void)ws_size;
  const float* spf = (const float*)d_in[0];
  const float* raw = (const float*)d_in[1];
  const int*   idx = (const int*)d_in[2];
  const int*   lab = (const int*)d_in[3];
  const int       S = in_sizes[0] / kD;
  const long long N = (long long)in_sizes[1] / kD;

  char* ws = (char*)d_ws;
  size_t off = 0;
  float*        fn    = (float*)(ws + off);        off += (size_t)S * kD * sizeof(float);
  float*        sums  = (float*)(ws + off);        off += (size_t)S * sizeof(float);
  unsigned int* cnts  = (unsigned int*)(ws + off); off += (size_t)S * sizeof(unsigned int);
  unsigned int* lcnt  = (unsigned int*)(ws + off); off += (size_t)S * kClasses * sizeof(unsigned int);
  int*          splab = (int*)(ws + off);          off += (size_t)S * sizeof(int);
  float*        acc   = (float*)(ws + off);        off += 8 * sizeof(float);

  // zero all accumulation state (sums .. acc), contiguous region after fn
  long long zwords = (long long)(((char*)(acc + 8)) - (char*)sums) / 4;
  k_zero<<<256, 256, 0, stream>>>((unsigned int*)sums, zwords);

  k_norm_super<<<(S + 255) / 256, 256, 0, stream>>>(spf, fn, acc, S);

  k_points<<<2048, 256, 0, stream>>>(raw, fn, idx, lab, sums, cnts, lcnt, N, S);

  k_sp_final<<<(S + 255) / 256, 256, 0, stream>>>(sums, cnts, lcnt, splab, acc, S);

  const int tiles = S >> 4;
  k_gram_wmma<<<tiles * tiles, 32, 0, stream>>>(fn, splab, acc, S);

  k_total<<<1, 32, 0, stream>>>(acc, (float*)d_out, S);
}